// GCNEncoder_6932077215862
// MI455X (gfx1250) — compile-verified
//
#include <hip/hip_runtime.h>
#include <hip/hip_bf16.h>

// GCN: out = D^-1/2 (A+I) D^-1/2 (X W) + b, 3 layers (relu, relu, none).
// Factored as: g = dinv[row] * (X @ W); agg = g (self loop) + scatter_add(g[src] -> agg[dst]);
// out = dinv[row] * agg + bias (+relu).

typedef float v2f __attribute__((ext_vector_type(2)));
typedef float v8f __attribute__((ext_vector_type(8)));

#define DIN 128   // K dimension for all three GEMMs

__device__ __forceinline__ void atomicAddF32(float* p, float v) {
    // lowers to global_atomic_add_f32 on gfx1250
    unsafeAtomicAdd(p, v);
}

// ---- degree / dinv -------------------------------------------------------

__global__ __launch_bounds__(256) void init_deg_kernel(float* deg, int n) {
    int i = blockIdx.x * blockDim.x + threadIdx.x;
    if (i < n) deg[i] = 1.0f;   // self-loop contributes 1 to every node
}

__global__ __launch_bounds__(256) void count_deg_kernel(const int* __restrict__ dst,
                                                        float* deg, int e) {
    int i = blockIdx.x * blockDim.x + threadIdx.x;
    if (i < e) atomicAddF32(&deg[dst[i]], 1.0f);
}

__global__ __launch_bounds__(256) void rsqrt_kernel(float* deg, int n) {
    int i = blockIdx.x * blockDim.x + threadIdx.x;
    if (i < n) deg[i] = rsqrtf(deg[i]);   // deg >= 1 always (self loops)
}

// ---- GEMM: g[m, n] = dinv[m] * sum_k A[m,k] * W[k,n] ---------------------
// One wave computes a 16-row strip across all NCOLS columns using
// v_wmma_f32_16x16x4_f32. K fixed at 128. M must be a multiple of 16
// (100000 = 6250*16). NCOLS is a compile-time constant (128 or 64) so the
// accumulators stay in VGPRs (no scratch spills).

template <int NCOLS>
__global__ __launch_bounds__(256)
void gemm_scale_kernel(const float* __restrict__ A, const float* __restrict__ W,
                       const float* __restrict__ dinv, float* __restrict__ g,
                       int M) {
    constexpr int NT = NCOLS / 16;        // 8 (d=128) or 4 (d=64) tiles
    const int lane    = threadIdx.x & 31;
    const int wave    = blockIdx.x * (blockDim.x >> 5) + (threadIdx.x >> 5);
    const int m0      = wave * 16;
    if (m0 >= M) return;                  // wave-uniform: EXEC stays all-ones
    const int halfSel = lane >> 4;        // 0: lanes 0-15, 1: lanes 16-31
    const int l16     = lane & 15;

    v8f acc[NT];
#pragma unroll
    for (int t = 0; t < NT; ++t) { v8f z = {}; acc[t] = z; }

    const float* arow = A + (size_t)(m0 + l16) * DIN;

#pragma unroll 4
    for (int k0 = 0; k0 < DIN; k0 += 4) {
        // A 16x4 frag: VGPR0 = K={k0 | k0+2}, VGPR1 = K={k0+1 | k0+3} by lane half
        const int kb = k0 + (halfSel << 1);
        v2f a = *(const v2f*)(arow + kb);
        const float* w0 = W + (size_t)kb * NCOLS + l16;
#pragma unroll
        for (int t = 0; t < NT; ++t) {
            v2f b;
            b.x = w0[t << 4];             // B row K=kb   (cols striped over lanes)
            b.y = w0[(t << 4) + NCOLS];   // B row K=kb+1
            acc[t] = __builtin_amdgcn_wmma_f32_16x16x4_f32(
                false, a, false, b, (short)0, acc[t], false, false);
        }
    }

    // C/D layout: VGPR r holds M = m0 + r + 8*halfSel, N = tile*16 + l16
    float scale[8];
#pragma unroll
    for (int r = 0; r < 8; ++r) scale[r] = dinv[m0 + r + (halfSel << 3)];

#pragma unroll
    for (int t = 0; t < NT; ++t) {
        const int col = (t << 4) + l16;
#pragma unroll
        for (int r = 0; r < 8; ++r) {
            const int grow = m0 + r + (halfSel << 3);
            g[(size_t)grow * NCOLS + col] = acc[t][r] * scale[r];
        }
    }
}

// ---- edge scatter-add: agg[dst] += g[src], d features per edge ------------
// One thread handles 4 consecutive floats; 2^csh threads (chunks) per edge.

__global__ __launch_bounds__(256)
void edge_agg_kernel(const float* __restrict__ g, float* __restrict__ agg,
                     const int* __restrict__ srcI, const int* __restrict__ dstI,
                     int E, int csh, int d) {
    const long long tid = (long long)blockIdx.x * blockDim.x + threadIdx.x;
    const int e = (int)(tid >> csh);
    if (e >= E) return;
    const int c = ((int)tid & ((1 << csh) - 1)) << 2;   // float offset in row
    const int s = srcI[e];
    const int t = dstI[e];
    const float4 v = *(const float4*)(g + (size_t)s * d + c);
    float* p = agg + (size_t)t * d + c;
    atomicAddF32(p + 0, v.x);
    atomicAddF32(p + 1, v.y);
    atomicAddF32(p + 2, v.z);
    atomicAddF32(p + 3, v.w);
}

// ---- finalize: out = (relu?)(dinv[i] * agg + bias) ------------------------

__global__ __launch_bounds__(256)
void finalize_kernel(const float* __restrict__ agg, const float* __restrict__ dinv,
                     const float* __restrict__ bias, float* __restrict__ out,
                     long long total, int dsh, int doRelu) {
    const long long idx = (long long)blockIdx.x * blockDim.x + threadIdx.x;
    if (idx >= total) return;
    const int i = (int)(idx >> dsh);
    const int f = (int)idx & ((1 << dsh) - 1);
    float v = dinv[i] * agg[idx] + bias[f];
    if (doRelu) v = fmaxf(v, 0.0f);
    out[idx] = v;
}

// ---- host-side orchestration ---------------------------------------------

static inline unsigned cdiv(long long a, long long b) {
    return (unsigned)((a + b - 1) / b);
}

extern "C" void kernel_launch(void* const* d_in, const int* in_sizes, int n_in,
                              void* d_out, int out_size, void* d_ws, size_t ws_size,
                              hipStream_t stream) {
    const float* x  = (const float*)d_in[0];
    const int*   ei = (const int*)d_in[1];
    const float* W1 = (const float*)d_in[2];
    const float* b1 = (const float*)d_in[3];
    const float* W2 = (const float*)d_in[4];
    const float* b2 = (const float*)d_in[5];
    const float* W3 = (const float*)d_in[6];
    const float* b3 = (const float*)d_in[7];

    const int N = in_sizes[0] / DIN;     // 100000
    const int E = in_sizes[1] / 2;       // 800000
    const int* srcI = ei;
    const int* dstI = ei + E;
    const int DH = 128, DO = 64;

    // workspace: dinv[N] | buf0[N*128] | buf1[N*128]  (~103 MB)
    float* dinv = (float*)d_ws;
    float* buf0 = dinv + N;
    float* buf1 = buf0 + (size_t)N * DH;

    const dim3 blk(256);
    const unsigned gemmGrid = cdiv((N + 15) / 16, 8);   // 8 waves per block

    // degrees -> dinv (in place)
    init_deg_kernel<<<cdiv(N, 256), blk, 0, stream>>>(dinv, N);
    count_deg_kernel<<<cdiv(E, 256), blk, 0, stream>>>(dstI, dinv, E);
    rsqrt_kernel<<<cdiv(N, 256), blk, 0, stream>>>(dinv, N);

    // ---- layer 1: x[N,128] @ W1[128,128] -----------------------------------
    gemm_scale_kernel<128><<<gemmGrid, blk, 0, stream>>>(x, W1, dinv, buf0, N);
    hipMemcpyAsync(buf1, buf0, (size_t)N * DH * sizeof(float),
                   hipMemcpyDeviceToDevice, stream);                    // self-loops
    edge_agg_kernel<<<cdiv((long long)E << 5, 256), blk, 0, stream>>>(
        buf0, buf1, srcI, dstI, E, 5, DH);
    finalize_kernel<<<cdiv((long long)N * DH, 256), blk, 0, stream>>>(
        buf1, dinv, b1, buf1, (long long)N * DH, 7, 1);

    // ---- layer 2: buf1[N,128] @ W2[128,128] --------------------------------
    gemm_scale_kernel<128><<<gemmGrid, blk, 0, stream>>>(buf1, W2, dinv, buf0, N);
    hipMemcpyAsync(buf1, buf0, (size_t)N * DH * sizeof(float),
                   hipMemcpyDeviceToDevice, stream);
    edge_agg_kernel<<<cdiv((long long)E << 5, 256), blk, 0, stream>>>(
        buf0, buf1, srcI, dstI, E, 5, DH);
    finalize_kernel<<<cdiv((long long)N * DH, 256), blk, 0, stream>>>(
        buf1, dinv, b2, buf1, (long long)N * DH, 7, 1);

    // ---- layer 3: buf1[N,128] @ W3[128,64] -> d_out ------------------------
    gemm_scale_kernel<64><<<gemmGrid, blk, 0, stream>>>(buf1, W3, dinv, buf0, N);
    hipMemcpyAsync(buf1, buf0, (size_t)N * DO * sizeof(float),
                   hipMemcpyDeviceToDevice, stream);
    edge_agg_kernel<<<cdiv((long long)E << 4, 256), blk, 0, stream>>>(
        buf0, buf1, srcI, dstI, E, 4, DO);
    finalize_kernel<<<cdiv((long long)N * DO, 256), blk, 0, stream>>>(
        buf1, dinv, b3, (float*)d_out, (long long)N * DO, 6, 0);
}